// ConstGateRNN_46136538694213
// MI455X (gfx1250) — compile-verified
//
#include <hip/hip_runtime.h>

// ConstGateRNN on MI455X (gfx1250, wave32).
// B=64,T=512,D=128,H=512. Outputs concat: y[B,T,1], hseq, gate, leak, rdiag (each [B,T,H]).
//
// Phase 1 (parallel): xw = x @ Wx^T  via fp32 WMMA, 65536 16x16 tiles -> workspace.
// Phase 2 (serial scan): 4 WGs x 16 batch rows; h state in LDS; 32 waves each owning a
// 16-wide H slice; per step one 16x16x512 fp32 WMMA tile (4 independent acc chains).
// A compile-time memory clobber in the t-loop stops LLVM from hoisting loop-invariant
// fragment loads into registers (which spilled under the 8-waves/SIMD ~128-VGPR cap).

typedef __attribute__((ext_vector_type(2))) float v2f;
typedef __attribute__((ext_vector_type(8))) float v8f;

#define NB 64
#define NT 512
#define ND 128
#define NH 512
#define S_GATE 0.7f
#define LDSH 516   // LDS row stride in floats (516 % 64 == 4 -> conflict-free column access)

#define WMMA4(ACC, AV, BV)                                                     \
  ACC = __builtin_amdgcn_wmma_f32_16x16x4_f32(false, AV, false, BV, (short)0,  \
                                              ACC, false, false)

// Grouped 16-K chunk: 8 loads first (clause-able), then 4 WMMAs.
#define K16_CHUNK(APTR, BPTR, K)                                               \
  {                                                                            \
    v2f a0_ = *(const v2f*)((APTR) + (K));                                     \
    v2f a1_ = *(const v2f*)((APTR) + (K) + 4);                                 \
    v2f a2_ = *(const v2f*)((APTR) + (K) + 8);                                 \
    v2f a3_ = *(const v2f*)((APTR) + (K) + 12);                                \
    v2f b0_ = *(const v2f*)((BPTR) + (K));                                     \
    v2f b1_ = *(const v2f*)((BPTR) + (K) + 4);                                 \
    v2f b2_ = *(const v2f*)((BPTR) + (K) + 8);                                 \
    v2f b3_ = *(const v2f*)((BPTR) + (K) + 12);                                \
    WMMA4(acc0, a0_, b0_);                                                     \
    WMMA4(acc1, a1_, b1_);                                                     \
    WMMA4(acc2, a2_, b2_);                                                     \
    WMMA4(acc3, a3_, b3_);                                                     \
  }

// ---------------------------------------------------------------------------
// Phase 1: xw[BT,H] = x[BT,D] @ Wx^T[D,H]   (x is [B,T,D] == [BT,D] contiguous)
// One wave per 16x16 output tile, K=128 -> 32 WMMAs, 8 waves per block.
// ---------------------------------------------------------------------------
__global__ __launch_bounds__(256)
void xw_gemm_kernel(const float* __restrict__ x,
                    const float* __restrict__ Wx,
                    float* __restrict__ xw)
{
  const int lane = threadIdx.x & 31;
  const int wave = threadIdx.x >> 5;
  const int half = lane >> 4;
  const int l16  = lane & 15;
  const int tile = blockIdx.x * 8 + wave;
  const int tn   = tile & (NH / 16 - 1);   // 0..31  (fast axis: reuse A rows in block)
  const int tm   = tile >> 5;              // 0..2047
  const int m0   = tm * 16, n0 = tn * 16;

  const float* Ap = x  + (size_t)(m0 + l16) * ND + 2 * half;  // A[m][k]
  const float* Bp = Wx + (size_t)(n0 + l16) * ND + 2 * half;  // B[k][n] = Wx[n][k]

  v8f acc0 = {}; v8f acc1 = {}; v8f acc2 = {}; v8f acc3 = {};
#pragma unroll
  for (int k0 = 0; k0 < ND; k0 += 16) {
    K16_CHUNK(Ap, Bp, k0);
  }
  v8f r = acc0 + acc1 + acc2 + acc3;
#pragma unroll
  for (int v = 0; v < 8; ++v)
    xw[(size_t)(m0 + v + 8 * half) * NH + n0 + l16] = r[v];
}

// ---------------------------------------------------------------------------
// Phase 2: the serial scan.
// ---------------------------------------------------------------------------
template <bool USE_XW>
__global__ __launch_bounds__(1024)
void constgate_rnn_kernel(const float* __restrict__ x,      // [B,T,D]
                          const float* __restrict__ Wx,     // [H,D]
                          const float* __restrict__ Wxb,    // [H]
                          const float* __restrict__ Wh,     // [H,H]
                          const float* __restrict__ outw,   // [H]
                          const float* __restrict__ outb,   // [1]
                          const float* __restrict__ xw,     // [BT,H] or null
                          float* __restrict__ y,            // [B,T]
                          float* __restrict__ hseq,         // [B,T,H]
                          float* __restrict__ rdiag)        // [B,T,H]
{
  __shared__ float lds_h[16 * LDSH];
  __shared__ float lds_ow[NH];

  const int tid  = threadIdx.x;
  const int lane = tid & 31;
  const int wave = tid >> 5;           // 0..31, owns H columns [wave*16, wave*16+16)
  const int half = lane >> 4;
  const int l16  = lane & 15;
  const int b0   = blockIdx.x * 16;
  const int n0   = wave * 16;
  const int ncol = n0 + l16;

  for (int i = tid; i < 16 * LDSH; i += blockDim.x) lds_h[i] = 0.0f;
  for (int i = tid; i < NH; i += blockDim.x) lds_ow[i] = outw[i];
  __syncthreads();

  const float wd   = Wh[(size_t)ncol * NH + ncol];
  const float bias = Wxb[ncol];
  const float ob   = outb[0];

  const float* xA  = x  + (size_t)(b0 + l16) * NT * ND + 2 * half;  // + t*ND + k0
  const float* wxB = Wx + (size_t)ncol * ND + 2 * half;             // + k0
  const float* whB = Wh + (size_t)ncol * NH + 2 * half;             // + k0
  const float* hA  = lds_h + l16 * LDSH + 2 * half;                 // + k0
  // Per-lane xw gather addresses (C/D layout rows), valid when USE_XW.
  const float* xwp = USE_XW ? (xw + (size_t)(b0 + 8 * half) * NT * NH + ncol) : nullptr;

  for (int t = 0; t < NT; ++t) {
    // Stop LLVM from hoisting loop-invariant fragment loads out of the t-loop:
    // under 32 waves/WGP the VGPR cap is ~128 and hoisting caused scratch spills.
    asm volatile("" ::: "memory");

    v8f acc0 = {}; v8f acc1 = {}; v8f acc2 = {}; v8f acc3 = {};
    float xwv[8];

    if (USE_XW) {
      // 8 scattered loads; issued up-front so they overlap the whole K-loop.
#pragma unroll
      for (int v = 0; v < 8; ++v)
        xwv[v] = xwp[((size_t)v * NT + t) * NH];
    } else {
      const float* xAt = xA + (size_t)t * ND;
      __builtin_prefetch(xAt + ND, 0, 3);   // next timestep's x row, WGP scope
#pragma unroll
      for (int k0 = 0; k0 < ND; k0 += 16) {
        K16_CHUNK(xAt, wxB, k0);
      }
    }

    // Recurrence: pre += h_prev[16,512] @ Wh^T[:,n0:n0+16]   (128 WMMAs, 4 chains)
#pragma unroll 4
    for (int k0 = 0; k0 < NH; k0 += 16) {
      K16_CHUNK(hA, whB, k0);
    }
    v8f pre = acc0 + acc1 + acc2 + acc3;

    float ht[8], hn[8];
#pragma unroll
    for (int v = 0; v < 8; ++v) {
      const int m = v + 8 * half;
      const float hprev = lds_h[m * LDSH + ncol];   // pre-update (writes are after barrier)
      float p = pre[v] + bias;
      if (USE_XW) p += xwv[v];
      const float th = tanhf(p);
      ht[v] = th;
      hn[v] = (1.0f - S_GATE) * hprev + S_GATE * th;
    }
    __syncthreads();   // all reads of old h complete

#pragma unroll
    for (int v = 0; v < 8; ++v) {
      const int m = v + 8 * half;
      lds_h[m * LDSH + ncol] = hn[v];
      const size_t g = ((size_t)(b0 + m) * NT + t) * NH + ncol;
      hseq[g]  = hn[v];
      rdiag[g] = S_GATE * (1.0f - ht[v] * ht[v]) * wd;
    }
    __syncthreads();   // new h visible

    // y[b,t] = h_new[b,:] . out_w + out_b — one wave per row, wave32 shfl reduce.
    if (wave < 16) {
      const int m = wave;
      float partial = 0.0f;
#pragma unroll
      for (int i = 0; i < NH / 32; ++i) {
        const int hc = lane + 32 * i;
        partial += lds_h[m * LDSH + hc] * lds_ow[hc];
      }
#pragma unroll
      for (int off = 16; off > 0; off >>= 1)
        partial += __shfl_xor(partial, off, 32);
      if (lane == 0)
        y[(size_t)(b0 + m) * NT + t] = partial + ob;
    }
    // No trailing barrier: next iteration only *reads* lds_h before its first barrier.
  }
}

// Pure-bandwidth fill of the constant gate (0.7) / leak (0.3) tensors (134 MB).
__global__ void fill_gate_leak_kernel(float4* __restrict__ gate,
                                      float4* __restrict__ leak,
                                      size_t n4)
{
  const float4 gv = make_float4(S_GATE, S_GATE, S_GATE, S_GATE);
  const float4 lv = make_float4(1.0f - S_GATE, 1.0f - S_GATE, 1.0f - S_GATE, 1.0f - S_GATE);
  size_t i  = (size_t)blockIdx.x * blockDim.x + threadIdx.x;
  size_t st = (size_t)gridDim.x * blockDim.x;
  for (; i < n4; i += st) {
    gate[i] = gv;
    leak[i] = lv;
  }
}

extern "C" void kernel_launch(void* const* d_in, const int* in_sizes, int n_in,
                              void* d_out, int out_size, void* d_ws, size_t ws_size,
                              hipStream_t stream) {
  (void)in_sizes; (void)n_in; (void)out_size;

  const float* x    = (const float*)d_in[0];
  const float* Wx   = (const float*)d_in[1];
  const float* Wxb  = (const float*)d_in[2];
  const float* Wh   = (const float*)d_in[3];
  const float* outw = (const float*)d_in[4];
  const float* outb = (const float*)d_in[5];

  float* out = (float*)d_out;
  const size_t BT  = (size_t)NB * NT;
  const size_t BTH = BT * NH;
  float* y     = out;
  float* hseq  = y + BT;
  float* gate  = hseq + BTH;
  float* leak  = gate + BTH;
  float* rdiag = leak + BTH;

  fill_gate_leak_kernel<<<2048, 256, 0, stream>>>((float4*)gate, (float4*)leak, BTH / 4);

  const bool use_ws = (ws_size >= BTH * sizeof(float));
  if (use_ws) {
    float* xw = (float*)d_ws;
    // 2048 x 32 tiles, 8 waves/block -> 8192 blocks: fills the whole chip.
    xw_gemm_kernel<<<(int)((BT / 16) * (NH / 16) / 8), 256, 0, stream>>>(x, Wx, xw);
    constgate_rnn_kernel<true><<<NB / 16, 1024, 0, stream>>>(
        x, Wx, Wxb, Wh, outw, outb, xw, y, hseq, rdiag);
  } else {
    constgate_rnn_kernel<false><<<NB / 16, 1024, 0, stream>>>(
        x, Wx, Wxb, Wh, outw, outb, nullptr, y, hseq, rdiag);
  }
}